// Net_10385230922599
// MI455X (gfx1250) — compile-verified
//
#include <hip/hip_runtime.h>

#define H 64
#define FIN 5

typedef __attribute__((ext_vector_type(2))) float v2f;
typedef __attribute__((ext_vector_type(8))) float v8f;

// ---------------------------------------------------------------- utilities

__global__ void zero_f32(float* p, long long cnt) {
  long long i = (long long)blockIdx.x * blockDim.x + threadIdx.x;
  if (i < cnt) p[i] = 0.0f;
}

// A[src, dst] += 1.0 into padded (ld x ld) buffer; exact & order-independent.
__global__ void scatter_ones(const int* __restrict__ ei, float* __restrict__ A,
                             int E, int ld) {
  int e = blockIdx.x * blockDim.x + threadIdx.x;
  if (e < E) {
    int s = ei[e], d = ei[E + e];
    atomicAdd(&A[(long long)s * ld + d], 1.0f);
  }
}

// Deterministic weighted scatter: Q8.24 fixed point via integer atomics.
__global__ void scatter_attr_fixed(const int* __restrict__ ei, const float* __restrict__ w,
                                   unsigned* __restrict__ A, int E, int ld) {
  int e = blockIdx.x * blockDim.x + threadIdx.x;
  if (e < E) {
    int s = ei[e], d = ei[E + e];
    unsigned fx = (unsigned)(w[e] * 16777216.0f + 0.5f);
    atomicAdd(&A[(long long)s * ld + d], fx);
  }
}

__global__ void fixed_to_float(unsigned* __restrict__ A, long long cnt) {
  long long i = (long long)blockIdx.x * blockDim.x + threadIdx.x;
  if (i < cnt) {
    float v = (float)A[i] * (1.0f / 16777216.0f);   // u32 0 -> 0.0f, pad stays zero
    ((float*)A)[i] = v;
  }
}

// BatchNorm1d training-mode (biased stats), one block per feature.
__global__ void bn_kernel(const float* __restrict__ x, const float* __restrict__ g,
                          const float* __restrict__ b, float* __restrict__ xn, int n) {
  __shared__ float ssum[256], ssq[256];
  int f = blockIdx.x;
  float s = 0.f, sq = 0.f;
  for (int i = threadIdx.x; i < n; i += blockDim.x) {
    float v = x[(long long)i * FIN + f];
    s += v; sq += v * v;
  }
  ssum[threadIdx.x] = s; ssq[threadIdx.x] = sq;
  __syncthreads();
  for (int st = 128; st > 0; st >>= 1) {
    if ((int)threadIdx.x < st) {
      ssum[threadIdx.x] += ssum[threadIdx.x + st];
      ssq[threadIdx.x]  += ssq[threadIdx.x + st];
    }
    __syncthreads();
  }
  float mu  = ssum[0] / (float)n;
  float var = ssq[0] / (float)n - mu * mu;
  float sc  = rsqrtf(var + 1e-5f) * g[f];
  float bb  = b[f];
  for (int i = threadIdx.x; i < n; i += blockDim.x)
    xn[(long long)i * FIN + f] = (x[(long long)i * FIN + f] - mu) * sc + bb;
}

// GCN normalization vectors over padded adjacency (np x np, zero pad):
// deg[d] = sum_s A[s,d] + selfloop_fix; dinv/dfix zero-padded to np.
__global__ void degree_kernel(const float* __restrict__ A, float* __restrict__ dinv,
                              float* __restrict__ dfix, int n, int np) {
  int d = blockIdx.x * blockDim.x + threadIdx.x;
  if (d >= np) return;
  if (d >= n) { dinv[d] = 0.0f; dfix[d] = 0.0f; return; }
  float diag = A[(long long)d * np + d];
  float fix  = (diag == 0.0f) ? 1.0f : 0.0f;
  float deg  = fix;
  for (int s = 0; s < n; ++s) deg += A[(long long)s * np + d];   // coalesced in d
  dinv[d] = (deg > 0.0f) ? rsqrtf(deg) : 0.0f;
  dfix[d] = fix;
}

// XWs = diag(dinv) * (X @ W), written into zero-padded (np x colp) buffer.
// Folds the source-side GCN normalization so the WMMA K-loop is a pure GEMM.
__global__ void xw_scaled_kernel(const float* __restrict__ X, const float* __restrict__ W,
                                 const float* __restrict__ dinv, float* __restrict__ Y,
                                 int n, int kdim, int cols, int colp, int np) {
  int idx = blockIdx.x * blockDim.x + threadIdx.x;
  if (idx >= np * colp) return;
  int r = idx / colp, c = idx % colp;
  float acc = 0.f;
  if (r < n && c < cols) {
    const float* xr = X + (long long)r * kdim;
    for (int t = 0; t < kdim; ++t) acc = fmaf(xr[t], W[(long long)t * cols + c], acc);
    acc *= dinv[r];
  }
  Y[idx] = acc;
}

__global__ void gelu_kernel(float* y, int cnt) {
  int i = blockIdx.x * blockDim.x + threadIdx.x;
  if (i < cnt) {
    float v = y[i];
    y[i] = 0.5f * v * (1.0f + erff(v * 0.70710678118654752f));
  }
}

__global__ void score_kernel(const float* __restrict__ h, const float* __restrict__ p,
                             float* __restrict__ score, int n) {
  int i = blockIdx.x * blockDim.x + threadIdx.x;
  if (i >= n) return;
  float nn = 0.f, s = 0.f;
#pragma unroll
  for (int j = 0; j < H; ++j) { nn += p[j] * p[j]; s += h[(long long)i * H + j] * p[j]; }
  score[i] = tanhf(s * rsqrtf(nn));
}

// Full descending bitonic sort of (score, index) in LDS; emit first k indices.
__global__ __launch_bounds__(1024)
void topk_kernel(const float* __restrict__ score, int* __restrict__ perm, int n, int k) {
  const int SZ = 4096;                        // n <= 3000
  __shared__ float key[SZ];
  __shared__ int   idx[SZ];
  for (int i = threadIdx.x; i < SZ; i += 1024) {
    bool ok = (i < n);
    key[i] = ok ? score[i] : -__builtin_inff();
    idx[i] = ok ? i : 0x7FFFFFFF;
  }
  __syncthreads();
  for (int size = 2; size <= SZ; size <<= 1) {
    for (int stride = size >> 1; stride > 0; stride >>= 1) {
      for (int t = threadIdx.x; t < SZ / 2; t += 1024) {
        int pos = 2 * t - (t & (stride - 1));
        int q   = pos + stride;
        float ka = key[pos], kb = key[q];
        int   ia = idx[pos], ib = idx[q];
        bool aAfter = (ka < kb) || (ka == kb && ia > ib);  // desc key, asc idx
        bool desc = ((pos & size) == 0);
        if (desc ? aAfter : !aAfter) {
          key[pos] = kb; key[q] = ka;
          idx[pos] = ib; idx[q] = ia;
        }
      }
      __syncthreads();
    }
  }
  for (int i = threadIdx.x; i < k; i += 1024) perm[i] = idx[i];
}

__global__ void gather_h(const float* __restrict__ h, const float* __restrict__ score,
                         const int* __restrict__ perm, float* __restrict__ hp, int k) {
  int i = blockIdx.x * blockDim.x + threadIdx.x;
  if (i >= k * H) return;
  int r = i / H, c = i % H;
  int src = perm[r];
  hp[i] = h[(long long)src * H + c] * score[src];
}

// Ap[kp x kp] = Asrc[perm x perm] with explicit zero padding (keeps invariant).
__global__ void gather_A(const float* __restrict__ Asrc, const int* __restrict__ perm,
                         float* __restrict__ Ap, int k, int npsrc, int kp) {
  int i = blockIdx.x * blockDim.x + threadIdx.x;
  if (i >= kp * kp) return;
  int r = i / kp, c = i % kp;
  Ap[i] = (r < k && c < k) ? Asrc[(long long)perm[r] * npsrc + perm[c]] : 0.0f;
}

__global__ void scatter_rows(const float* __restrict__ hsrc, const int* __restrict__ perm,
                             float* __restrict__ u, int k) {
  int i = blockIdx.x * blockDim.x + threadIdx.x;
  if (i >= k * H) return;
  int r = i / H, c = i % H;
  u[(long long)perm[r] * H + c] = hsrc[i];
}

__global__ void concat_kernel(const float* __restrict__ res, const float* __restrict__ u,
                              float* __restrict__ hc, int n) {
  int i = blockIdx.x * blockDim.x + threadIdx.x;
  if (i >= n * 2 * H) return;
  int r = i / (2 * H), c = i % (2 * H);
  hc[i] = (c < H) ? res[(long long)r * H + c] : u[(long long)r * H + (c - H)];
}

// ------------------------------------------------- WMMA fp32 matmul kernels
//
// V_WMMA_F32_16X16X4_F32 layouts (ISA 7.12.2, wave32):
//   A 16x4:  lane&15 = M row; VGPR t holds K = (lane>>4)*2 + t
//   B 4x16:  lane&15 = N col; VGPR t holds K = (lane>>4)*2 + t
//   C/D:     VGPR j -> row j + (lane>>4)*8, col = lane&15
// All operands come from zero-padded np x np / np x colp buffers: the K-loops
// have no bounds checks, no exec manipulation, no compares.

// GCN aggregation: out[d,h] = bias[h] + dinv[d]*( sum_s A[s,d]*XWs[s,h]
//                                                 + dfix[d]*XWs[d,h] )
// (XWs already carries dinv[s]; self-loop fix applied at the store.)
template <int NT>
__global__ __launch_bounds__(32)
void gcn_aggregate_wmma(const float* __restrict__ A, const float* __restrict__ dinv,
                        const float* __restrict__ dfix, const float* __restrict__ XWs,
                        const float* __restrict__ bias, float* __restrict__ out,
                        int n, int np, int cols, int colp) {
  int lane = threadIdx.x;
  int mr = lane & 15;
  int kb = (lane >> 4) << 1;
  int d0 = blockIdx.x << 4;
  int d  = d0 + mr;

  v8f zero = {0.f,0.f,0.f,0.f,0.f,0.f,0.f,0.f};
  v8f acc[NT];
#pragma unroll
  for (int t = 0; t < NT; ++t) acc[t] = zero;

  const float* Ac = A + d;                    // walk column d of A
  for (int s0 = 0; s0 < np; s0 += 4) {
    int s = s0 + kb;
    v2f a;
    a[0] = Ac[(long long)s * np];             // A^T tile: rows=d, K=s
    a[1] = Ac[(long long)(s + 1) * np];
#pragma unroll
    for (int nt = 0; nt < NT; ++nt) {
      int hc = (nt << 4) + mr;
      v2f bm;
      bm[0] = XWs[(long long)s * colp + hc];
      bm[1] = XWs[(long long)(s + 1) * colp + hc];
      acc[nt] = __builtin_amdgcn_wmma_f32_16x16x4_f32(
          false, a, false, bm, (short)0, acc[nt], false, false);
    }
  }
#pragma unroll
  for (int nt = 0; nt < NT; ++nt) {
    int hc = (nt << 4) + mr;
    if (hc >= cols) continue;
    float bv = bias[hc];
#pragma unroll
    for (int j = 0; j < 8; ++j) {
      int row = d0 + j + ((lane >> 4) << 3);
      if (row < n) {
        float self = dfix[row] * XWs[(long long)row * colp + hc];
        out[(long long)row * cols + hc] = dinv[row] * (acc[nt][j] + self) + bv;
      }
    }
  }
}

// augment_adj: (A+I)@(A+I) with zeroed diagonal  ==  offdiag(A@A + 2A).
// Pure A@A in the K-loop (row-side tiles load contiguous K pairs as b64);
// +2A and the diagonal mask are applied at the store. 64x32 C tile per wave.
__global__ __launch_bounds__(32)
void augment_wmma(const float* __restrict__ A, float* __restrict__ out, int np) {
  int lane = threadIdx.x;
  int mr = lane & 15;
  int kb = (lane >> 4) << 1;
  int i0 = blockIdx.x << 6;   // 64 rows
  int j0 = blockIdx.y << 5;   // 32 cols

  v8f zero = {0.f,0.f,0.f,0.f,0.f,0.f,0.f,0.f};
  v8f acc[4][2];
#pragma unroll
  for (int rt = 0; rt < 4; ++rt)
#pragma unroll
    for (int ct = 0; ct < 2; ++ct) acc[rt][ct] = zero;

  for (int k0 = 0; k0 < np; k0 += 4) {
    v2f a[4];
#pragma unroll
    for (int rt = 0; rt < 4; ++rt) {
      int i = i0 + (rt << 4) + mr;
      a[rt] = *(const v2f*)(A + (long long)i * np + k0 + kb);   // b64, 8B aligned
    }
    int s = k0 + kb;
    v2f b[2];
#pragma unroll
    for (int ct = 0; ct < 2; ++ct) {
      int j = j0 + (ct << 4) + mr;
      b[ct][0] = A[(long long)s * np + j];
      b[ct][1] = A[(long long)(s + 1) * np + j];
    }
#pragma unroll
    for (int rt = 0; rt < 4; ++rt)
#pragma unroll
      for (int ct = 0; ct < 2; ++ct)
        acc[rt][ct] = __builtin_amdgcn_wmma_f32_16x16x4_f32(
            false, a[rt], false, b[ct], (short)0, acc[rt][ct], false, false);
  }
#pragma unroll
  for (int rt = 0; rt < 4; ++rt) {
#pragma unroll
    for (int ct = 0; ct < 2; ++ct) {
      int j = j0 + (ct << 4) + mr;
#pragma unroll
      for (int jj = 0; jj < 8; ++jj) {
        int row = i0 + (rt << 4) + jj + ((lane >> 4) << 3);
        long long o = (long long)row * np + j;
        float v = acc[rt][ct][jj] + 2.0f * A[o];
        out[o] = (row == j) ? 0.0f : v;   // pad rows/cols stay exactly zero
      }
    }
  }
}

// ------------------------------------------------------------------- driver

extern "C" void kernel_launch(void* const* d_in, const int* in_sizes, int n_in,
                              void* d_out, int out_size, void* d_ws, size_t ws_size,
                              hipStream_t stream) {
  (void)n_in; (void)out_size; (void)ws_size;
  const float* x   = (const float*)d_in[0];
  const int*   ei  = (const int*)  d_in[1];
  const float* ea  = (const float*)d_in[2];
  const float* bng = (const float*)d_in[3];
  const float* bnb = (const float*)d_in[4];
  const float* dw0 = (const float*)d_in[5];
  const float* db0 = (const float*)d_in[6];
  const float* dw_all[3] = {(const float*)d_in[7],  (const float*)d_in[9],  (const float*)d_in[11]};
  const float* db_all[3] = {(const float*)d_in[8],  (const float*)d_in[10], (const float*)d_in[12]};
  const float* pv[3]     = {(const float*)d_in[13], (const float*)d_in[14], (const float*)d_in[15]};
  const float* uw1 = (const float*)d_in[16];
  const float* ub1 = (const float*)d_in[17];
  const float* uw2 = (const float*)d_in[18];
  const float* ub2 = (const float*)d_in[19];
  const float* uw3 = (const float*)d_in[20];
  const float* ub3 = (const float*)d_in[21];
  const float* fw  = (const float*)d_in[22];
  const float* fb  = (const float*)d_in[23];

  const int N  = in_sizes[0] / FIN;   // 3000
  const int E  = in_sizes[2];         // 30000
  const int n1 = (N + 1) / 2;         // 1500
  const int n2 = (n1 + 1) / 2;        // 750
  const int n3 = (n2 + 1) / 2;        // 375
  auto pad64 = [](int v) { return (v + 63) & ~63; };
  const int np0 = pad64(N);           // 3008
  const int np1 = pad64(n1);          // 1536
  const int np2 = pad64(n2);          // 768
  const int np3 = pad64(n3);          // 384

  // workspace layout
  float* ws = (float*)d_ws;
  long long off = 0;
  auto alloc = [&](long long cnt) -> float* { float* p = ws + off; off += cnt; return p; };
  float* A0p    = alloc((long long)np0 * np0);   // raw adjacency (kept for up path)
  float* Aaugp  = alloc((long long)np0 * np0);   // augment scratch; reused for A_attr
  float* A1p    = alloc((long long)np1 * np1);
  float* A2p    = alloc((long long)np2 * np2);
  float* A3p    = alloc((long long)np3 * np3);
  float* xn     = alloc((long long)N * FIN);
  float* xwb    = alloc((long long)np0 * H);     // padded, dinv-prescaled XW
  float* ha     = alloc((long long)N * H);
  float* hb     = alloc((long long)N * H);
  float* xs0    = alloc((long long)N * H);
  float* xs1    = alloc((long long)n1 * H);
  float* xs2    = alloc((long long)n2 * H);
  float* hcat   = alloc((long long)N * 2 * H);
  float* scoreb = alloc(N);
  float* dinv   = alloc(np0);
  float* dfix   = alloc(np0);
  int* perm1 = (int*)alloc(n1);
  int* perm2 = (int*)alloc(n2);
  int* perm3 = (int*)alloc(n3);
  int* perm_all[3] = {perm1, perm2, perm3};

  const int TB = 256;
  auto cdiv = [](int a, int b) { return (a + b - 1) / b; };

  auto run_gcn = [&](const float* A, int n, int np, const float* X, int kdim,
                     const float* W, int cols, const float* bias, float* out) {
    int colp = (cols > 16) ? 64 : 16;
    degree_kernel<<<cdiv(np, TB), TB, 0, stream>>>(A, dinv, dfix, n, np);
    xw_scaled_kernel<<<cdiv(np * colp, TB), TB, 0, stream>>>(X, W, dinv, xwb,
                                                             n, kdim, cols, colp, np);
    if (cols > 16)
      gcn_aggregate_wmma<4><<<np >> 4, 32, 0, stream>>>(A, dinv, dfix, xwb, bias,
                                                        out, n, np, cols, colp);
    else
      gcn_aggregate_wmma<1><<<np >> 4, 32, 0, stream>>>(A, dinv, dfix, xwb, bias,
                                                        out, n, np, cols, colp);
  };

  // ---- BatchNorm1d
  bn_kernel<<<FIN, 256, 0, stream>>>(x, bng, bnb, xn, N);

  // ---- dense adjacency A0 (weights = 1), zero-padded
  zero_f32<<<cdiv(np0 * np0, TB), TB, 0, stream>>>(A0p, (long long)np0 * np0);
  scatter_ones<<<cdiv(E, TB), TB, 0, stream>>>(ei, A0p, E, np0);

  // ---- first GCN + gelu, save skip xs0
  run_gcn(A0p, N, np0, xn, FIN, dw0, H, db0, ha);
  gelu_kernel<<<cdiv(N * H, TB), TB, 0, stream>>>(ha, N * H);
  hipMemcpyAsync(xs0, ha, sizeof(float) * (size_t)N * H, hipMemcpyDeviceToDevice, stream);

  // ---- down path
  const float* Acur = A0p;
  float* Apool[3] = {A1p, A2p, A3p};
  float* xsave[3] = {xs1, xs2, 0};
  int nlv[4]  = {N, n1, n2, n3};
  int nplv[4] = {np0, np1, np2, np3};
  for (int lvl = 0; lvl < 3; ++lvl) {
    int n = nlv[lvl], k = nlv[lvl + 1];
    int np = nplv[lvl], kp = nplv[lvl + 1];
    augment_wmma<<<dim3(np >> 6, np >> 5), 32, 0, stream>>>(Acur, Aaugp, np);
    score_kernel<<<cdiv(n, TB), TB, 0, stream>>>(ha, pv[lvl], scoreb, n);
    topk_kernel<<<1, 1024, 0, stream>>>(scoreb, perm_all[lvl], n, k);
    gather_h<<<cdiv(k * H, TB), TB, 0, stream>>>(ha, scoreb, perm_all[lvl], hb, k);
    gather_A<<<cdiv(kp * kp, TB), TB, 0, stream>>>(Aaugp, perm_all[lvl], Apool[lvl], k, np, kp);
    run_gcn(Apool[lvl], k, kp, hb, H, dw_all[lvl], H, db_all[lvl], ha);
    gelu_kernel<<<cdiv(k * H, TB), TB, 0, stream>>>(ha, k * H);
    if (lvl < 2)
      hipMemcpyAsync(xsave[lvl], ha, sizeof(float) * (size_t)k * H,
                     hipMemcpyDeviceToDevice, stream);
    Acur = Apool[lvl];
  }

  // ---- up path (sum_res=False: concat skip)
  struct Up { const float* res; const float* A; int n; int np; int k; int* perm;
              const float* W; const float* b; bool act; };
  Up ups[3] = {
    { xs2, A2p, n2, np2, n3, perm3, uw1, ub1, true  },
    { xs1, A1p, n1, np1, n2, perm2, uw2, ub2, true  },
    { xs0, A0p, N,  np0, n1, perm1, uw3, ub3, false },
  };
  for (int i = 0; i < 3; ++i) {
    Up& u = ups[i];
    zero_f32<<<cdiv(u.n * H, TB), TB, 0, stream>>>(hb, (long long)u.n * H);
    scatter_rows<<<cdiv(u.k * H, TB), TB, 0, stream>>>(ha, u.perm, hb, u.k);
    concat_kernel<<<cdiv(u.n * 2 * H, TB), TB, 0, stream>>>(u.res, hb, hcat, u.n);
    run_gcn(u.A, u.n, u.np, hcat, 2 * H, u.W, H, u.b, ha);
    if (u.act) gelu_kernel<<<cdiv(u.n * H, TB), TB, 0, stream>>>(ha, u.n * H);
  }

  // ---- final GCN with edge_attr-weighted adjacency (deterministic fixed-point build)
  zero_f32<<<cdiv(np0 * np0, TB), TB, 0, stream>>>(Aaugp, (long long)np0 * np0);
  scatter_attr_fixed<<<cdiv(E, TB), TB, 0, stream>>>(ei, ea, (unsigned*)Aaugp, E, np0);
  fixed_to_float<<<cdiv(np0 * np0, TB), TB, 0, stream>>>((unsigned*)Aaugp, (long long)np0 * np0);
  run_gcn(Aaugp, N, np0, ha, H, fw, 3, fb, (float*)d_out);
}